// HungarianMatcher_6047313953236
// MI455X (gfx1250) — compile-verified
//
#include <hip/hip_runtime.h>
#include <hip/hip_bf16.h>

typedef __attribute__((ext_vector_type(2))) float v2f;
typedef __attribute__((ext_vector_type(8))) float v8f;

#define HW   25600   // 160*160
#define BNUM 8
#define NQ   100
#define BN   800     // B*N
#define TT   160     // total targets
#define KCLS 80
#define MPER 20

#define MASK_COST 2.0f
#define ALPHA     0.25f
#define EPS_F     1e-8f
#define DICE_EPS  1e-4f

// ---------------------------------------------------------------------------
// Row sums-of-squares for pred_masks (rows 0..799) and target_masks (800..959)
// ---------------------------------------------------------------------------
__global__ void __launch_bounds__(256)
norms_kernel(const float* __restrict__ pm, const float* __restrict__ tmm,
             float* __restrict__ pnorm, float* __restrict__ tnorm) {
  const int row = blockIdx.x;
  const float* src = (row < BN) ? pm + (size_t)row * HW
                                : tmm + (size_t)(row - BN) * HW;
  const float4* s4 = reinterpret_cast<const float4*>(src);
  float s = 0.f;
  for (int i = threadIdx.x; i < HW / 4; i += 256) {
    float4 v = s4[i];
    s += v.x * v.x + v.y * v.y + v.z * v.z + v.w * v.w;
  }
  __shared__ float red[256];
  red[threadIdx.x] = s;
  __syncthreads();
  for (int off = 128; off > 0; off >>= 1) {
    if (threadIdx.x < off) red[threadIdx.x] += red[threadIdx.x + off];
    __syncthreads();
  }
  if (threadIdx.x == 0) {
    if (row < BN) pnorm[row] = red[0];
    else          tnorm[row - BN] = red[0];
  }
}

// ---------------------------------------------------------------------------
// Async global->LDS staging: one instruction moves 32 lanes x 16B = 512B.
// 8 instructions = one 16x64-float tile half-pair per matrix.
// Tracked with ASYNCcnt; no VGPR round-trip.
// ---------------------------------------------------------------------------
__device__ __forceinline__ void async_tile_load(unsigned lds, const float* g) {
  unsigned long long ga = (unsigned long long)(size_t)g;
#pragma unroll
  for (int j = 0; j < 8; ++j) {
    asm volatile("global_load_async_to_lds_b128 %0, %1, off offset:%2"
                 :: "v"(lds), "v"(ga), "i"(j * 16) : "memory");
  }
}

// ---------------------------------------------------------------------------
// dot = pm (800 x 25600) @ tm^T (25600 x 160) via V_WMMA_F32_16X16X4_F32.
// One wave per 16x16 output tile; K double-buffered through padded LDS
// (row stride 68 floats -> conflict-free ds_load_2addr_b64 fragment loads),
// filled by GLOBAL_LOAD_ASYNC_TO_LDS_B128 and synchronized with
// s_wait_asynccnt only (no barriers, no VGPR staging).
// ---------------------------------------------------------------------------
__global__ void __launch_bounds__(32)
gemm_wmma_kernel(const float* __restrict__ pm, const float* __restrict__ tmm,
                 float* __restrict__ dot) {
  __shared__ float As[2][16][68];
  __shared__ float Bs[2][16][68];

  const int tile = blockIdx.x;        // 0..499
  const int tM   = tile / 10;         // 0..49
  const int tN   = tile - tM * 10;    // 0..9
  const int lane = threadIdx.x;       // 0..31 (wave32)
  const int m0 = tM * 16, n0 = tN * 16;

  // 2 lanes per row, 32 floats (8 x b128) each
  const int lrow = lane >> 1;
  const int lcol = (lane & 1) * 32;
  const float* aSrc = pm  + (size_t)(m0 + lrow) * HW + lcol;
  const float* bSrc = tmm + (size_t)(n0 + lrow) * HW + lcol;

  // per-lane LDS byte offsets (low 32 bits of generic = wave-relative LDS addr)
  const unsigned ldsA0 = (unsigned)(size_t)&As[0][lrow][lcol];
  const unsigned ldsA1 = (unsigned)(size_t)&As[1][lrow][lcol];
  const unsigned ldsB0 = (unsigned)(size_t)&Bs[0][lrow][lcol];
  const unsigned ldsB1 = (unsigned)(size_t)&Bs[1][lrow][lcol];

  // fragment addressing per ISA 16x4 fp32 A layout:
  //   lanes 0-15 hold K={0,1}, lanes 16-31 hold K={2,3}; B symmetric.
  const int fm   = lane & 15;
  const int koff = (lane < 16) ? 0 : 2;

  v8f acc = {};

  // prologue: fill buffer 0 with K[0..64)
  async_tile_load(ldsA0, aSrc);
  async_tile_load(ldsB0, bSrc);

  for (int k0 = 0; k0 < HW; k0 += 128) {
    // prefetch buffer 1 with K[k0+64 .. k0+128) (always in range: HW % 128 == 0)
    async_tile_load(ldsA1, aSrc + k0 + 64);
    async_tile_load(ldsB1, bSrc + k0 + 64);
    // 16 just issued outstanding -> waiting <=16 means buffer 0 is complete
    asm volatile("s_wait_asynccnt 16" ::: "memory");
#pragma unroll
    for (int ks = 0; ks < 64; ks += 4) {
      v2f a = *reinterpret_cast<const v2f*>(&As[0][fm][ks + koff]);
      v2f b = *reinterpret_cast<const v2f*>(&Bs[0][fm][ks + koff]);
      // 8 args: (neg_a, A, neg_b, B, c_mod, C, reuse_a, reuse_b)
      acc = __builtin_amdgcn_wmma_f32_16x16x4_f32(
          false, a, false, b, (short)0, acc, false, false);
    }
    if (k0 + 128 < HW) {
      // prefetch buffer 0 with next chunk
      async_tile_load(ldsA0, aSrc + k0 + 128);
      async_tile_load(ldsB0, bSrc + k0 + 128);
      asm volatile("s_wait_asynccnt 16" ::: "memory");
    } else {
      asm volatile("s_wait_asynccnt 0" ::: "memory");
    }
#pragma unroll
    for (int ks = 0; ks < 64; ks += 4) {
      v2f a = *reinterpret_cast<const v2f*>(&As[1][fm][ks + koff]);
      v2f b = *reinterpret_cast<const v2f*>(&Bs[1][fm][ks + koff]);
      acc = __builtin_amdgcn_wmma_f32_16x16x4_f32(
          false, a, false, b, (short)0, acc, false, false);
    }
  }

  // C/D layout: VGPR r -> M=r (lanes 0-15), M=r+8 (lanes 16-31); N = lane%16
#pragma unroll
  for (int r = 0; r < 8; ++r) {
    int Mr = (lane < 16) ? r : (r + 8);
    int Nc = lane & 15;
    dot[(size_t)(m0 + Mr) * TT + n0 + Nc] = acc[r];
  }
}

// ---------------------------------------------------------------------------
// C[bn, t] = MASK_COST * mask_score + focal classification cost, nan_to_num.
// ---------------------------------------------------------------------------
__global__ void __launch_bounds__(256)
combine_kernel(const float* __restrict__ dot, const float* __restrict__ pnorm,
               const float* __restrict__ tnorm, const float* __restrict__ logits,
               const int* __restrict__ tgt_ids, float* __restrict__ C) {
  const int idx = blockIdx.x * 256 + threadIdx.x;  // 0..127999
  if (idx >= BN * TT) return;
  const int bn = idx / TT;
  const int t  = idx - bn * TT;

  const int cls = tgt_ids[t];
  const float x = logits[bn * KCLS + cls];
  const float p = 1.f / (1.f + expf(-x));
  const float omp = 1.f - p;
  // GAMMA == 2 -> p^gamma is a multiply
  const float pos = ALPHA * omp * omp * (-logf(p + EPS_F));
  const float neg = (1.f - ALPHA) * p * p * (-logf(omp + EPS_F));
  const float cc = pos - neg;

  const float num = 2.f * dot[idx] + DICE_EPS;
  const float den = pnorm[bn] + tnorm[t] + DICE_EPS;
  float v = -MASK_COST * (num / den) + cc;
  if (!isfinite(v)) v = 0.f;
  C[idx] = v;
}

// ---------------------------------------------------------------------------
// Per-image Jonker-Volgenant assignment on the 100x20 block slice,
// transposed to 20x100 exactly like the reference (float64 potentials).
// One scalar thread per image.
// ---------------------------------------------------------------------------
__global__ void lsa_kernel(const float* __restrict__ C,
                           float* __restrict__ out_rows,
                           float* __restrict__ out_cols) {
  const int img = blockIdx.x;  // 0..7
  const int n = MPER;          // 20 (rows after transpose)
  const int m = NQ;            // 100 (cols after transpose)
  // cost[t][pred] = C[img, pred, img*MPER + t]
  const float* base = C + (size_t)img * NQ * TT + (size_t)img * MPER;

  double u[MPER + 1], v[NQ + 1], minv[NQ + 1];
  int p[NQ + 1], way[NQ + 1];
  bool used[NQ + 1];
  const double INF = 1e18;

  for (int i = 0; i <= n; ++i) u[i] = 0.0;
  for (int j = 0; j <= m; ++j) { v[j] = 0.0; p[j] = 0; way[j] = 0; }

  for (int i = 1; i <= n; ++i) {
    p[0] = i;
    int j0 = 0;
    for (int j = 0; j <= m; ++j) { minv[j] = INF; used[j] = false; }
    while (true) {
      used[j0] = true;
      const int i0 = p[j0];
      double delta = INF;
      int j1 = -1;
      for (int j = 1; j <= m; ++j) {
        if (!used[j]) {
          double cur = (double)base[(size_t)(j - 1) * TT + (i0 - 1)] - u[i0] - v[j];
          if (cur < minv[j]) { minv[j] = cur; way[j] = j0; }
          if (minv[j] < delta) { delta = minv[j]; j1 = j; }  // first-min tie-break
        }
      }
      for (int j = 0; j <= m; ++j) {
        if (used[j]) { u[p[j]] += delta; v[j] -= delta; }
        else         { minv[j] -= delta; }
      }
      j0 = j1;
      if (p[j0] == 0) break;
    }
    while (true) {
      const int j1 = way[j0];
      p[j0] = p[j1];
      j0 = j1;
      if (j0 == 0) break;
    }
  }

  // j ascending -> pred index (row after un-transpose) already sorted
  int cnt = 0;
  for (int j = 1; j <= m; ++j) {
    if (p[j] != 0) {
      out_rows[img * MPER + cnt] = (float)(j - 1);      // pred index
      out_cols[img * MPER + cnt] = (float)(p[j] - 1);   // target index
      ++cnt;
    }
  }
}

// ---------------------------------------------------------------------------
extern "C" void kernel_launch(void* const* d_in, const int* in_sizes, int n_in,
                              void* d_out, int out_size, void* d_ws, size_t ws_size,
                              hipStream_t stream) {
  (void)in_sizes; (void)n_in; (void)out_size; (void)ws_size;
  const float* pred_masks   = (const float*)d_in[0];  // (8,100,160,160)
  const float* pred_logits  = (const float*)d_in[1];  // (8,100,80)
  const float* target_masks = (const float*)d_in[2];  // (160,160,160)
  const int*   tgt_ids      = (const int*)d_in[3];    // (160,)

  float* C        = (float*)d_out;                    // 128000
  float* out_rows = C + BN * TT;                      // 160
  float* out_cols = out_rows + BNUM * MPER;           // 160

  float* dot   = (float*)d_ws;                        // 800*160
  float* pnorm = dot + (size_t)BN * TT;               // 800
  float* tnorm = pnorm + BN;                          // 160

  norms_kernel<<<BN + TT, 256, 0, stream>>>(pred_masks, target_masks, pnorm, tnorm);
  gemm_wmma_kernel<<<(BN / 16) * (TT / 16), 32, 0, stream>>>(pred_masks, target_masks, dot);
  combine_kernel<<<(BN * TT) / 256, 256, 0, stream>>>(dot, pnorm, tnorm,
                                                      pred_logits, tgt_ids, C);
  lsa_kernel<<<BNUM, 1, 0, stream>>>(C, out_rows, out_cols);
}